// _GATConvSparse_57363583205823
// MI455X (gfx1250) — compile-verified
//
#include <hip/hip_runtime.h>
#include <hip/hip_bf16.h>
#include <math.h>

typedef __attribute__((ext_vector_type(2))) float v2f;
typedef __attribute__((ext_vector_type(8))) float v8f;

#define IN_DIM 128
#define OUT_DIM 64

// ---------------------------------------------------------------------------
// Kernel 1: Wh = h @ W^T via V_WMMA_F32_16X16X4_F32.
// One wave (32 lanes) computes a 16-row x 64-col tile of Wh.
// A fragment (16x4 f32): lanes 0-15 hold M=0..15 K={0,1}; lanes 16-31 K={2,3}.
// B fragment (4x16 f32): lanes 0-15 hold N=0..15 K={0,1}; lanes 16-31 K={2,3}.
// C/D (16x16 f32): VGPR g -> M = g + 8*(lane>=16), N = lane&15.
// ---------------------------------------------------------------------------
__global__ void gat_gemm_wmma_kernel(const float* __restrict__ h,
                                     const float* __restrict__ W,   // (64,128) row-major
                                     float* __restrict__ Wh,        // (N,64)
                                     int n_nodes) {
    const int lane = threadIdx.x & 31;
    const int wave = threadIdx.x >> 5;
    const int m0 = (blockIdx.x * (blockDim.x >> 5) + wave) * 16;
    if (m0 >= n_nodes) return;                       // wave-uniform: EXEC stays all-1s

    const int m  = lane & 15;
    const int kb = (lane >> 4) << 1;                 // 0 or 2
    const float* __restrict__ arow = h + (size_t)(m0 + m) * IN_DIM;
    const float* __restrict__ b0r  = W + (size_t)(m)      * IN_DIM; // n-tile 0 row
    const float* __restrict__ b1r  = W + (size_t)(16 + m) * IN_DIM;
    const float* __restrict__ b2r  = W + (size_t)(32 + m) * IN_DIM;
    const float* __restrict__ b3r  = W + (size_t)(48 + m) * IN_DIM;

    v8f c0 = {}, c1 = {}, c2 = {}, c3 = {};
    #pragma unroll
    for (int kt = 0; kt < IN_DIM / 4; ++kt) {
        const int k = kt * 4 + kb;
        v2f a  = *(const v2f*)(arow + k);
        v2f b0 = *(const v2f*)(b0r + k);
        v2f b1 = *(const v2f*)(b1r + k);
        v2f b2 = *(const v2f*)(b2r + k);
        v2f b3 = *(const v2f*)(b3r + k);
        c0 = __builtin_amdgcn_wmma_f32_16x16x4_f32(false, a, false, b0, (short)0, c0, false, false);
        c1 = __builtin_amdgcn_wmma_f32_16x16x4_f32(false, a, false, b1, (short)0, c1, false, false);
        c2 = __builtin_amdgcn_wmma_f32_16x16x4_f32(false, a, false, b2, (short)0, c2, false, false);
        c3 = __builtin_amdgcn_wmma_f32_16x16x4_f32(false, a, false, b3, (short)0, c3, false, false);
    }

    const int row_base = m0 + ((lane >> 4) << 3);
    const int col = lane & 15;
    #pragma unroll
    for (int g = 0; g < 8; ++g) {
        float* p = Wh + (size_t)(row_base + g) * OUT_DIM + col;
        p[0]  = c0[g];
        p[16] = c1[g];
        p[32] = c2[g];
        p[48] = c3[g];
    }
}

// ---------------------------------------------------------------------------
// Kernel 2: per-node attention scores s1[n]=Wh[n].a[:64], s2[n]=Wh[n].a[64:]
// ---------------------------------------------------------------------------
__global__ void node_scores_kernel(const float* __restrict__ Wh,
                                   const float* __restrict__ a,
                                   float* __restrict__ s1,
                                   float* __restrict__ s2,
                                   int n_nodes) {
    const int nid = blockIdx.x * blockDim.x + threadIdx.x;
    if (nid >= n_nodes) return;
    const float4* __restrict__ wv = (const float4*)(Wh + (size_t)nid * OUT_DIM);
    const float4* __restrict__ a1 = (const float4*)a;
    const float4* __restrict__ a2 = (const float4*)(a + OUT_DIM);
    float d1 = 0.f, d2 = 0.f;
    #pragma unroll
    for (int j = 0; j < OUT_DIM / 4; ++j) {
        float4 w = wv[j], x = a1[j], y = a2[j];
        d1 += w.x * x.x + w.y * x.y + w.z * x.z + w.w * x.w;
        d2 += w.x * y.x + w.y * y.y + w.z * y.z + w.w * y.w;
    }
    s1[nid] = d1;
    s2[nid] = d2;
}

// ---------------------------------------------------------------------------
// Kernel 3: zero-init accumulators (out, alpha_node, denom, maxbits)
// ---------------------------------------------------------------------------
__global__ void init_kernel(float* __restrict__ out, float* __restrict__ alpha_node,
                            float* __restrict__ denom, unsigned* __restrict__ maxbits,
                            int n_nodes) {
    const int i = blockIdx.x * blockDim.x + threadIdx.x;
    const int total = n_nodes * OUT_DIM;
    if (i < total) out[i] = 0.f;
    if (i < n_nodes) { alpha_node[i] = 0.f; denom[i] = 0.f; }
    if (i == 0) maxbits[0] = 0u;   // maps below every real float in flipped encoding
}

// ---------------------------------------------------------------------------
// Kernel 4: alpha = (s1[src] + s2[dst]) * ew ; segment-sum onto dst
// ---------------------------------------------------------------------------
__global__ void edge_alpha_kernel(const int* __restrict__ src, const int* __restrict__ dst,
                                  const float* __restrict__ ew,
                                  const float* __restrict__ s1, const float* __restrict__ s2,
                                  float* __restrict__ alpha_node, int n_edges) {
    const int e = blockIdx.x * blockDim.x + threadIdx.x;
    if (e >= n_edges) return;
    const int s = src[e], d = dst[e];
    const float alpha = (s1[s] + s2[d]) * ew[e];
    atomicAdd(&alpha_node[d], alpha);
}

// Order-preserving float->uint mapping for atomicMax
__device__ __forceinline__ unsigned fmap(float f) {
    unsigned b = __float_as_uint(f);
    return (b & 0x80000000u) ? ~b : (b | 0x80000000u);
}
__device__ __forceinline__ float funmap(unsigned u) {
    unsigned b = (u & 0x80000000u) ? (u & 0x7FFFFFFFu) : ~u;
    return __uint_as_float(b);
}

// ---------------------------------------------------------------------------
// Kernel 5: global max of alpha_node (block reduce + atomicMax on mapped bits)
// ---------------------------------------------------------------------------
__global__ void max_reduce_kernel(const float* __restrict__ alpha_node,
                                  unsigned* __restrict__ maxbits, int n_nodes) {
    __shared__ float red[256];
    const int i = blockIdx.x * blockDim.x + threadIdx.x;
    float v = (i < n_nodes) ? alpha_node[i] : -INFINITY;
    red[threadIdx.x] = v;
    __syncthreads();
    for (int s = 128; s > 0; s >>= 1) {
        if (threadIdx.x < s) red[threadIdx.x] = fmaxf(red[threadIdx.x], red[threadIdx.x + s]);
        __syncthreads();
    }
    if (threadIdx.x == 0) atomicMax(maxbits, fmap(red[0]));
}

// ---------------------------------------------------------------------------
// Kernel 6: alpha_exp = exp(alpha_node - max)
// ---------------------------------------------------------------------------
__global__ void exp_kernel(const float* __restrict__ alpha_node,
                           const unsigned* __restrict__ maxbits,
                           float* __restrict__ alpha_exp, int n_nodes) {
    const int i = blockIdx.x * blockDim.x + threadIdx.x;
    if (i >= n_nodes) return;
    const float mx = funmap(maxbits[0]);
    alpha_exp[i] = __expf(alpha_node[i] - mx);
}

// ---------------------------------------------------------------------------
// Kernel 7: denom[dst] += alpha_exp[src]
// ---------------------------------------------------------------------------
__global__ void edge_denom_kernel(const int* __restrict__ src, const int* __restrict__ dst,
                                  const float* __restrict__ alpha_exp,
                                  float* __restrict__ denom, int n_edges) {
    const int e = blockIdx.x * blockDim.x + threadIdx.x;
    if (e >= n_edges) return;
    atomicAdd(&denom[dst[e]], alpha_exp[src[e]]);
}

// ---------------------------------------------------------------------------
// Kernel 8: out[dst] += (alpha_exp[src]/(denom[dst]+1e-9)) * Wh[src]
// 16 threads per edge, each handles 4 contiguous features (float4 gather).
// ---------------------------------------------------------------------------
__global__ void edge_scatter_kernel(const int* __restrict__ src, const int* __restrict__ dst,
                                    const float* __restrict__ alpha_exp,
                                    const float* __restrict__ denom,
                                    const float* __restrict__ Wh,
                                    float* __restrict__ out, int n_edges) {
    const long long t = (long long)blockIdx.x * blockDim.x + threadIdx.x;
    const long long e = t >> 4;
    if (e >= n_edges) return;
    const int q = ((int)t & 15) << 2;        // feature offset 0,4,...,60
    const int s = src[e], d = dst[e];
    const float coef = alpha_exp[s] / (denom[d] + 1e-9f);
    const float4 w = *(const float4*)(Wh + (size_t)s * OUT_DIM + q);
    float* o = out + (size_t)d * OUT_DIM + q;
    atomicAdd(o + 0, coef * w.x);
    atomicAdd(o + 1, coef * w.y);
    atomicAdd(o + 2, coef * w.z);
    atomicAdd(o + 3, coef * w.w);
}

// ---------------------------------------------------------------------------
// Kernel 9: in-place ELU
// ---------------------------------------------------------------------------
__global__ void elu_kernel(float* __restrict__ out, int total) {
    const int i = blockIdx.x * blockDim.x + threadIdx.x;
    if (i >= total) return;
    const float x = out[i];
    out[i] = (x > 0.f) ? x : (__expf(x) - 1.f);
}

extern "C" void kernel_launch(void* const* d_in, const int* in_sizes, int n_in,
                              void* d_out, int out_size, void* d_ws, size_t ws_size,
                              hipStream_t stream) {
    const float* h  = (const float*)d_in[0];
    const int*   ei = (const int*)d_in[1];
    const float* ew = (const float*)d_in[2];
    const float* W  = (const float*)d_in[3];
    const float* av = (const float*)d_in[4];

    const int n_nodes = in_sizes[0] / IN_DIM;
    const int n_edges = in_sizes[2];
    const int* src = ei;
    const int* dst = ei + n_edges;

    // workspace layout
    float* Wh         = (float*)d_ws;                     // n_nodes * 64
    float* s1         = Wh + (size_t)n_nodes * OUT_DIM;   // n_nodes
    float* s2         = s1 + n_nodes;                     // n_nodes
    float* alpha_node = s2 + n_nodes;                     // n_nodes
    float* alpha_exp  = alpha_node + n_nodes;             // n_nodes
    float* denom      = alpha_exp + n_nodes;              // n_nodes
    unsigned* maxbits = (unsigned*)(denom + n_nodes);     // 1

    float* out = (float*)d_out;

    const int B = 256;

    // 1) WMMA GEMM: 8 waves/block, 16 rows per wave
    {
        const int waves  = (n_nodes + 15) / 16;
        const int blocks = (waves + 7) / 8;
        gat_gemm_wmma_kernel<<<blocks, 256, 0, stream>>>(h, W, Wh, n_nodes);
    }
    // 2) per-node scores
    node_scores_kernel<<<(n_nodes + B - 1) / B, B, 0, stream>>>(Wh, av, s1, s2, n_nodes);
    // 3) init accumulators
    {
        const int total = n_nodes * OUT_DIM;
        init_kernel<<<(total + B - 1) / B, B, 0, stream>>>(out, alpha_node, denom, maxbits, n_nodes);
    }
    // 4) alpha segment-sum onto dst
    edge_alpha_kernel<<<(n_edges + B - 1) / B, B, 0, stream>>>(src, dst, ew, s1, s2,
                                                              alpha_node, n_edges);
    // 5) global max
    max_reduce_kernel<<<(n_nodes + B - 1) / B, B, 0, stream>>>(alpha_node, maxbits, n_nodes);
    // 6) exp
    exp_kernel<<<(n_nodes + B - 1) / B, B, 0, stream>>>(alpha_node, maxbits, alpha_exp, n_nodes);
    // 7) denom segment-sum
    edge_denom_kernel<<<(n_edges + B - 1) / B, B, 0, stream>>>(src, dst, alpha_exp, denom, n_edges);
    // 8) weighted feature scatter
    {
        const long long total = (long long)n_edges * 16;
        const int blocks = (int)((total + B - 1) / B);
        edge_scatter_kernel<<<blocks, B, 0, stream>>>(src, dst, alpha_exp, denom, Wh, out, n_edges);
    }
    // 9) ELU
    {
        const int total = n_nodes * OUT_DIM;
        elu_kernel<<<(total + B - 1) / B, B, 0, stream>>>(out, total);
    }
}